// TopKRouter_83176336654411
// MI455X (gfx1250) — compile-verified
//
#include <hip/hip_runtime.h>
#include <hip/hip_bf16.h>
#include <math.h>

// MoE top-2 router for MI455X (gfx1250, wave32).
// logits = HS[32768,2048] @ W[64,2048]^T (fp32), softmax, top-2, renorm.
// Bandwidth-bound (AI = 32 FLOP/B; HBM floor ~12us at 23.3 TB/s). fp32 WMMA
// (V_WMMA_F32_16X16X4_F32) keeps reference numerics while lifting compute off
// the VALU. Global->LDS staging uses GLOBAL_LOAD_ASYNC_TO_LDS_B128 (ASYNCcnt)
// with double-buffered LDS so copies of chunk i+1 overlap WMMAs on chunk i.

typedef float v2f __attribute__((ext_vector_type(2)));
typedef float v8f __attribute__((ext_vector_type(8)));

#define TOKENS 32768
#define DMODEL 2048
#define NEXP   64
#define KC     64            // K-chunk staged through LDS
#define NCH    (DMODEL / KC) // 32 chunks
#define LDSP   68            // padded LDS row stride (floats): 272B, 16B-aligned
#define BM     32            // tokens per block

__device__ __forceinline__ void async_copy_b128(unsigned lds_off, const float* gptr) {
    // global -> LDS direct copy, tracked by ASYNCcnt (no VGPR round-trip).
    asm volatile("global_load_async_to_lds_b128 %0, %1, off"
                 :: "v"(lds_off), "v"((unsigned long long)(size_t)gptr)
                 : "memory");
}

__device__ __forceinline__ void wait_async0() {
    asm volatile("s_wait_asynccnt 0x0" ::: "memory");
}

__global__ __launch_bounds__(256)
void topk_router_kernel(const float* __restrict__ hs,
                        const float* __restrict__ w,
                        float* __restrict__ out_logits,
                        float* __restrict__ out_prob,
                        int*   __restrict__ out_idx) {
    __shared__ float lds_a[2][BM   * LDSP];  // 32 tokens  x KC, double-buffered
    __shared__ float lds_b[2][NEXP * LDSP];  // 64 experts x KC, double-buffered
    __shared__ float lds_c[BM * LDSP];       // 32 tokens  x 64 logits (staging)

    const int tid  = threadIdx.x;
    const int lane = tid & 31;
    const int wv   = tid >> 5;     // wave 0..7
    const int rt   = wv >> 2;      // token row-tile 0..1  (16 tokens each)
    const int ct   = wv & 3;       // expert col-tile 0..3 (16 experts each)
    const int tok0 = blockIdx.x * BM;

    // Per-thread cooperative-copy coordinates (float4 granularity).
    // A chunk: 32x64 f32 = 512 x b128, 2 per thread. B chunk: 64x64 = 1024, 4/thread.
    const int arow0 = tid >> 4,           acol = (tid & 15) << 2;
    const int arow1 = (tid + 256) >> 4;
    const int brow0 = tid >> 4;           // rows 0..15
    const int brow1 = (tid + 256) >> 4;   // 16..31
    const int brow2 = (tid + 512) >> 4;   // 32..47
    const int brow3 = (tid + 768) >> 4;   // 48..63

    auto issue_chunk = [&](int kb, int bi) {
        async_copy_b128((unsigned)(size_t)&lds_a[bi][arow0 * LDSP + acol],
                        hs + (size_t)(tok0 + arow0) * DMODEL + kb + acol);
        async_copy_b128((unsigned)(size_t)&lds_a[bi][arow1 * LDSP + acol],
                        hs + (size_t)(tok0 + arow1) * DMODEL + kb + acol);
        async_copy_b128((unsigned)(size_t)&lds_b[bi][brow0 * LDSP + acol],
                        w + (size_t)brow0 * DMODEL + kb + acol);
        async_copy_b128((unsigned)(size_t)&lds_b[bi][brow1 * LDSP + acol],
                        w + (size_t)brow1 * DMODEL + kb + acol);
        async_copy_b128((unsigned)(size_t)&lds_b[bi][brow2 * LDSP + acol],
                        w + (size_t)brow2 * DMODEL + kb + acol);
        async_copy_b128((unsigned)(size_t)&lds_b[bi][brow3 * LDSP + acol],
                        w + (size_t)brow3 * DMODEL + kb + acol);
    };

    // Per-lane WMMA fragment offsets (ISA §7.12.2):
    // A 16x4 f32: lane l -> row M=(l&15), K-offset 2*(l>>4), float2 = VGPR0/1
    // B 4x16 f32: lane l -> col N=(l&15) (expert row in lds_b), same K pattern
    const int aFragOff = ((lane & 15) + 16 * rt) * LDSP + ((lane >> 4) << 1);
    const int bFragOff = ((lane & 15) + 16 * ct) * LDSP + ((lane >> 4) << 1);

    // Prologue: fill buffer 0 with chunk 0.
    issue_chunk(0, 0);
    wait_async0();
    __syncthreads();

    v8f acc = {};

    for (int i = 0; i < NCH; ++i) {
        const int bi = i & 1;
        // Kick off next chunk's copies into the other buffer (overlaps WMMAs).
        if (i + 1 < NCH) issue_chunk((i + 1) * KC, bi ^ 1);
        // L2 hint two chunks ahead (global_prefetch_b8).
        if (i + 2 < NCH)
            __builtin_prefetch(hs + (size_t)(tok0 + arow0) * DMODEL + (i + 2) * KC + acol, 0, 0);

        const float* aBase = &lds_a[bi][aFragOff];
        const float* bBase = &lds_b[bi][bFragOff];
        #pragma unroll
        for (int k0 = 0; k0 < KC; k0 += 4) {
            v2f af = *(const v2f*)(aBase + k0);
            v2f bf = *(const v2f*)(bBase + k0);
            acc = __builtin_amdgcn_wmma_f32_16x16x4_f32(
                /*neg_a=*/false, af, /*neg_b=*/false, bf,
                /*c_mod=*/(short)0, acc, /*reuse_a=*/false, /*reuse_b=*/false);
        }

        wait_async0();     // this wave's next-chunk copies have landed in LDS
        __syncthreads();   // everyone's copies have landed; reads of bi complete
    }

    // ---- stage C tile to LDS: VGPR v -> (M = v + 8*(lane>>4), N = lane&15) ----
    {
        const int n  = lane & 15;
        const int mh = (lane >> 4) << 3;
        #pragma unroll
        for (int v = 0; v < 8; ++v)
            lds_c[((16 * rt + mh + v) * LDSP) + 16 * ct + n] = acc[v];
    }
    __syncthreads();

    // ---- logits out: 32x64 block contiguous in global; 2 float4/thread ----
    #pragma unroll
    for (int j = 0; j < 2; ++j) {
        const int i   = tid + j * 256;
        const int row = i >> 4;
        const int c4  = (i & 15) << 2;
        float4 val = *(const float4*)&lds_c[row * LDSP + c4];
        *(float4*)(out_logits + (size_t)(tok0 + row) * NEXP + c4) = val;
    }

    // ---- top-2 + renormalized softmax (softmax Z cancels exactly) ----
    if (tid < BM) {
        const float* lr = &lds_c[tid * LDSP];
        float b1v = -3.402823466e+38f, b2v = -3.402823466e+38f;
        int i1 = 0, i2 = 0;
        for (int e = 0; e < NEXP; ++e) {
            float v = lr[e];
            if (v > b1v)      { b2v = b1v; i2 = i1; b1v = v; i1 = e; }
            else if (v > b2v) { b2v = v; i2 = e; }
        }
        const float e2    = expf(b2v - b1v);
        const float denom = 1.0f + e2;
        const int t = tok0 + tid;
        out_prob[t * 2 + 0] = 1.0f / denom;
        out_prob[t * 2 + 1] = e2 / denom;
        out_idx[t * 2 + 0]  = i1;
        out_idx[t * 2 + 1]  = i2;
    }
}

extern "C" void kernel_launch(void* const* d_in, const int* in_sizes, int n_in,
                              void* d_out, int out_size, void* d_ws, size_t ws_size,
                              hipStream_t stream) {
    (void)in_sizes; (void)n_in; (void)out_size; (void)d_ws; (void)ws_size;
    const float* hs = (const float*)d_in[0];
    const float* w  = (const float*)d_in[1];
    float* out        = (float*)d_out;
    float* out_logits = out;                                        // [32768*64]
    float* out_prob   = out + (size_t)TOKENS * NEXP;                // [32768*2]
    int*   out_idx    = (int*)(out + (size_t)TOKENS * NEXP
                                   + (size_t)TOKENS * 2);           // [32768*2]
    dim3 grid(TOKENS / BM);  // 1024 blocks
    topk_router_kernel<<<grid, 256, 0, stream>>>(hs, w, out_logits, out_prob, out_idx);
}